// SimpleMaskModule_2671469658671
// MI455X (gfx1250) — compile-verified
//
#include <hip/hip_runtime.h>
#include <cstdint>

// Problem constants (match reference: lambdas is (2048, 512) fp32)
#define RDIM 2048
#define VDIM 512
#define TILE 64          // output tile edge per workgroup
#define VC   128         // V-chunk staged in LDS per iteration
#define LSTR (VC + 4)    // 132-float LDS row stride: 16B-aligned rows, spreads banks

// delta[i,j] = -sum_v |lam[i,v]-lam[j,v]|;  mask = exp(delta)
// Symmetric: only upper-triangle blocks compute; each writes tile + transpose.
// 256 threads = 16x16 thread grid, each thread owns a 4x4 output sub-tile.
__global__ __launch_bounds__(256) void pairwise_l1_exp_kernel(
    const float* __restrict__ lam, float* __restrict__ mask)
{
    const int i0 = blockIdx.y * TILE;
    const int j0 = blockIdx.x * TILE;
    if (j0 < i0) return;   // symmetry: lower triangle produced by transposed store

    __shared__ float As[TILE * LSTR];   // rows i0..i0+63, VC floats each
    __shared__ float Bs[TILE * LSTR];   // rows j0..j0+63, VC floats each

    const int tid = threadIdx.x;
    const int tx  = tid & 15;           // output-column group
    const int ty  = tid >> 4;           // output-row group

    float acc[4][4];
#pragma unroll
    for (int r = 0; r < 4; ++r)
#pragma unroll
        for (int c = 0; c < 4; ++c) acc[r][c] = 0.0f;

    for (int vb = 0; vb < VDIM; vb += VC) {
        __syncthreads();  // previous chunk's compute finished before LDS overwrite

        // Stage 64x128 floats (2048 float4) per matrix with CDNA5 async DMA:
        // 8 float4 slots per thread per matrix.
#pragma unroll
        for (int k = 0; k < 8; ++k) {
            const int f   = tid + k * 256;     // float4 slot 0..2047
            const int row = f >> 5;            // 32 float4 per staged row
            const int q   = (f & 31) << 2;     // float offset within row
            const float* ga = lam + (size_t)(i0 + row) * VDIM + vb + q;
            const float* gb = lam + (size_t)(j0 + row) * VDIM + vb + q;
            uint32_t la = (uint32_t)(uintptr_t)&As[row * LSTR + q];
            uint32_t lb = (uint32_t)(uintptr_t)&Bs[row * LSTR + q];
            asm volatile("global_load_async_to_lds_b128 %0, %1, off"
                         :: "v"(la), "v"(ga) : "memory");
            asm volatile("global_load_async_to_lds_b128 %0, %1, off"
                         :: "v"(lb), "v"(gb) : "memory");
        }
        // Wait for this wave's async transfers (ASYNCcnt), then workgroup barrier
        // so every wave sees all staged data.
        asm volatile("s_wait_asynccnt 0" ::: "memory");
        __syncthreads();

        const float* ap = &As[(ty * 4) * LSTR];
        const float* bp = &Bs[(tx * 4) * LSTR];

#pragma unroll 4
        for (int v = 0; v < VC; v += 4) {
            float4 a[4], b[4];
#pragma unroll
            for (int r = 0; r < 4; ++r)
                a[r] = *(const float4*)(ap + r * LSTR + v);
#pragma unroll
            for (int c = 0; c < 4; ++c)
                b[c] = *(const float4*)(bp + c * LSTR + v);
#pragma unroll
            for (int r = 0; r < 4; ++r)
#pragma unroll
                for (int c = 0; c < 4; ++c) {
                    // v_sub_f32 + v_add_f32 with |src| modifier per component
                    acc[r][c] += __builtin_fabsf(a[r].x - b[c].x)
                               + __builtin_fabsf(a[r].y - b[c].y)
                               + __builtin_fabsf(a[r].z - b[c].z)
                               + __builtin_fabsf(a[r].w - b[c].w);
                }
        }
    }

    // e[r][c] = exp(-acc[r][c])  (v_exp_f32; 16 per thread, negligible)
    float e[4][4];
#pragma unroll
    for (int r = 0; r < 4; ++r)
#pragma unroll
        for (int c = 0; c < 4; ++c)
            e[r][c] = __expf(-acc[r][c]);

    // Direct tile: mask[i, j0+tx*4 .. +3], coalesced float4 stores
#pragma unroll
    for (int r = 0; r < 4; ++r) {
        const int i = i0 + ty * 4 + r;
        float4 o = make_float4(e[r][0], e[r][1], e[r][2], e[r][3]);
        *(float4*)&mask[(size_t)i * RDIM + j0 + tx * 4] = o;
    }

    // Transposed tile: mask[j, i0+ty*4 .. +3] (symmetry). For diagonal blocks
    // this re-writes identical bit patterns — benign and deterministic.
#pragma unroll
    for (int c = 0; c < 4; ++c) {
        const int j = j0 + tx * 4 + c;
        float4 o = make_float4(e[0][c], e[1][c], e[2][c], e[3][c]);
        *(float4*)&mask[(size_t)j * RDIM + i0 + ty * 4] = o;
    }
}

// Second tuple output: lambdas passthrough copy into the tail of d_out.
__global__ void copy_f4_kernel(const float4* __restrict__ s,
                               float4* __restrict__ d, int n)
{
    int idx = blockIdx.x * blockDim.x + threadIdx.x;
    if (idx < n) d[idx] = s[idx];
}

extern "C" void kernel_launch(void* const* d_in, const int* in_sizes, int n_in,
                              void* d_out, int out_size, void* d_ws, size_t ws_size,
                              hipStream_t stream)
{
    (void)in_sizes; (void)n_in; (void)out_size; (void)d_ws; (void)ws_size;

    const float* lam = (const float*)d_in[0];
    float* out = (float*)d_out;

    dim3 grid(RDIM / TILE, RDIM / TILE);   // 32 x 32 tiles; lower triangle exits
    pairwise_l1_exp_kernel<<<grid, dim3(256), 0, stream>>>(lam, out);

    const int n4 = (RDIM * VDIM) / 4;      // 262144 float4
    copy_f4_kernel<<<(n4 + 255) / 256, 256, 0, stream>>>(
        (const float4*)lam, (float4*)(out + (size_t)RDIM * RDIM), n4);
}